// CausalPerformerAttention_8418135900415
// MI455X (gfx1250) — compile-verified
//
#include <hip/hip_runtime.h>
#include <math.h>

// ---------------------------------------------------------------------------
// CDNA5 (gfx1250) wave32 WMMA f32 16x16x4 helpers
// ---------------------------------------------------------------------------
typedef float v2f __attribute__((ext_vector_type(2)));
typedef float v8f __attribute__((ext_vector_type(8)));

__device__ __forceinline__ v8f wmma4(v2f a, v2f b, v8f c) {
  // D = A(16x4) * B(4x16) + C(16x16), all f32.
  return __builtin_amdgcn_wmma_f32_16x16x4_f32(
      /*neg_a=*/false, a, /*neg_b=*/false, b,
      /*c_mod=*/(short)0, c, /*reuse_a=*/false, /*reuse_b=*/false);
}

// Problem constants
#define Bq   2
#define Tq   1024
#define Cq   768
#define Hq   12
#define Dq   64
#define Fq   64
#define BH   (Bq * Hq)        // 24
#define ROWS (Bq * Tq)        // 2048
#define QKVC (3 * Cq)         // 2304

// ---------------------------------------------------------------------------
// Kernel 1/4: out[M,N] = A[M,K] @ W[N,K]^T    (x@W_attn.T and att@W_proj.T)
// 256 threads = 8 waves; block tile 64x128; wave tile 16x64 (4 accumulators).
// LDS double-buffered K-chunks (KC=16), coalesced float4 global loads staged
// through registers so they overlap WMMA on the other buffer.
// ---------------------------------------------------------------------------
#define KC  16
#define LDP 20    // padded LDS row stride (floats): 16B-aligned, conflict-free

__global__ __launch_bounds__(256) void gemm_xwT_kernel(
    const float* __restrict__ A, const float* __restrict__ W,
    float* __restrict__ out, int M, int N, int K) {
  __shared__ float As[2][64 * LDP];
  __shared__ float Bs[2][128 * LDP];

  const int tid  = threadIdx.x;
  const int wave = tid >> 5;
  const int lane = tid & 31;
  const int half = lane >> 4;
  const int l16  = lane & 15;
  const int m0 = blockIdx.x * 64;
  const int n0 = blockIdx.y * 128;
  const int wm = (wave & 3) * 16;   // wave row offset in block tile
  const int wn = (wave >> 2) * 64;  // wave col offset in block tile

  // staging map: 4 threads per row cover 16 cols; 256 threads = 64 rows/pass
  const int srow = tid >> 2;        // 0..63
  const int scol = (tid & 3) * 4;   // 0,4,8,12
  const float* Abase = A + (size_t)m0 * K;
  const float* Bbase = W + (size_t)n0 * K;

  float4 a0, b0, b1;
  // ---- prologue: chunk 0 -> buffer 0 ----
  a0 = *(const float4*)(Abase + (size_t)srow * K + scol);
  b0 = *(const float4*)(Bbase + (size_t)srow * K + scol);
  b1 = *(const float4*)(Bbase + (size_t)(srow + 64) * K + scol);
  *(float4*)&As[0][srow * LDP + scol]        = a0;
  *(float4*)&Bs[0][srow * LDP + scol]        = b0;
  *(float4*)&Bs[0][(srow + 64) * LDP + scol] = b1;
  __syncthreads();

  v8f acc[4] = {};
  const int nk = K / KC;
  for (int kc = 0; kc < nk; ++kc) {
    const int cur = kc & 1;
    if (kc + 1 < nk) {  // fetch next chunk into registers (overlaps compute)
      const int k0 = (kc + 1) * KC;
      a0 = *(const float4*)(Abase + (size_t)srow * K + k0 + scol);
      b0 = *(const float4*)(Bbase + (size_t)srow * K + k0 + scol);
      b1 = *(const float4*)(Bbase + (size_t)(srow + 64) * K + k0 + scol);
      if (kc + 2 < nk) {  // gfx1250 global_prefetch of the chunk after that
        __builtin_prefetch(Abase + (size_t)srow * K + k0 + KC + scol, 0, 1);
        __builtin_prefetch(Bbase + (size_t)srow * K + k0 + KC + scol, 0, 1);
      }
    }
#pragma unroll
    for (int ks = 0; ks < KC; ks += 4) {
      const int kk2 = ks + 2 * half;
      v2f a = *(const v2f*)&As[cur][(wm + l16) * LDP + kk2];
#pragma unroll
      for (int nt = 0; nt < 4; ++nt) {
        v2f b = *(const v2f*)&Bs[cur][(wn + nt * 16 + l16) * LDP + kk2];
        acc[nt] = wmma4(a, b, acc[nt]);
      }
    }
    if (kc + 1 < nk) {
      const int nxt = cur ^ 1;
      *(float4*)&As[nxt][srow * LDP + scol]        = a0;
      *(float4*)&Bs[nxt][srow * LDP + scol]        = b0;
      *(float4*)&Bs[nxt][(srow + 64) * LDP + scol] = b1;
    }
    __syncthreads();
  }

#pragma unroll
  for (int nt = 0; nt < 4; ++nt)
#pragma unroll
    for (int r = 0; r < 8; ++r) {
      const int m = m0 + wm + r + 8 * half;
      out[(size_t)m * N + n0 + wn + nt * 16 + l16] = acc[nt][r];
    }
}

// ---------------------------------------------------------------------------
// Kernel 2: phi features. Stages the 16x64 head-row panel in LDS (coalesced),
// computes row norms from LDS, then x_proj = panel @ proj via WMMA and
// exp(x_proj - 0.5*norm)/8.  128 threads = 4 waves, one 16x16 tile per wave.
// ---------------------------------------------------------------------------
#define LDQ 68   // padded stride for 64-float rows (16B aligned, no conflicts)

__global__ __launch_bounds__(128) void feature_kernel(
    const float* __restrict__ qkv, const float* __restrict__ proj,
    float* __restrict__ qp, float* __restrict__ kp) {
  __shared__ float Ablk[16 * LDQ];
  __shared__ float nrm[16];

  const int which = blockIdx.y;                 // 0 = q, 1 = k
  const int bh = blockIdx.x / (Tq / 16);
  const int t0 = (blockIdx.x % (Tq / 16)) * 16;
  const int b = bh / Hq, h = bh % Hq;
  const int tid = threadIdx.x;
  const int wave = tid >> 5, lane = tid & 31;
  const int half = lane >> 4, l16 = lane & 15;
  const int n0 = wave * 16;

  {  // stage: 8 threads per row, 8 floats each
    const int r = tid >> 3, c = (tid & 7) * 8;
    const float* src = qkv + (size_t)(b * Tq + t0 + r) * QKVC + which * Cq + h * Dq;
    *(float4*)&Ablk[r * LDQ + c]     = *(const float4*)(src + c);
    *(float4*)&Ablk[r * LDQ + c + 4] = *(const float4*)(src + c + 4);
  }
  __syncthreads();
  if (tid < 16) {
    float s = 0.f;
    const float* row = &Ablk[tid * LDQ];
#pragma unroll 8
    for (int d = 0; d < Dq; ++d) s += row[d] * row[d];
    nrm[tid] = s;
  }

  v8f acc = {};
  for (int k = 0; k < Dq; k += 4) {
    const int kk = k + 2 * half;
    v2f a = *(const v2f*)&Ablk[l16 * LDQ + kk];
    v2f bb; bb.x = proj[kk * Fq + n0 + l16];        // coalesced 16-lane rows
            bb.y = proj[(kk + 1) * Fq + n0 + l16];
    acc = wmma4(a, bb, acc);
  }
  __syncthreads();  // nrm ready for all waves

  float* o = which ? kp : qp;
#pragma unroll
  for (int r = 0; r < 8; ++r) {
    const int mm = r + 8 * half;
    const float val = expf(acc[r] - 0.5f * nrm[mm]) * 0.125f; // 1/sqrt(64)
    o[(size_t)(bh * Tq + t0 + mm) * Fq + n0 + l16] = val;
  }
}

// ---------------------------------------------------------------------------
// Kernel 3: chunked causal linear-attention scan.  One block per (b,h),
// 128 threads = 4 waves.  State S(64x64)+z(64) resident in LDS.  Per chunk
// of 16 timesteps (all fragments from staged LDS tiles):
//   M    = tril(Q' K'^T)                 (16x16, K=64)
//   num  = Q' @ S_prev + M @ V           (16x64)
//   den  = Q'.z_prev + rowsum(M) + eps
//   S   += K'^T @ V ;  z += colsum(K')
// ---------------------------------------------------------------------------
#define LDM 18   // padded stride for 16-float rows of Mloc

__global__ __launch_bounds__(128) void scan_kernel(
    const float* __restrict__ qp, const float* __restrict__ kp,
    const float* __restrict__ qkv, float* __restrict__ att) {
  __shared__ float S[Fq * Dq];          // 16 KB state
  __shared__ float Qc[16 * LDQ];
  __shared__ float Kc[16 * LDQ];
  __shared__ float Vc[16 * LDQ];
  __shared__ float Mloc[16 * LDM];
  __shared__ float zden[Fq + 16];       // [0..63]=z, [64..79]=denominator

  const int bh = blockIdx.x;
  const int b = bh / Hq, h = bh % Hq;
  const int tid = threadIdx.x;
  const int wave = tid >> 5, lane = tid & 31;
  const int half = lane >> 4, l16 = lane & 15;
  const int n0 = wave * 16;             // this wave's d-tile

  const float* qpb = qp + (size_t)bh * Tq * Fq;
  const float* kpb = kp + (size_t)bh * Tq * Fq;
  const float* vb  = qkv + (size_t)b * Tq * QKVC + 2 * Cq + h * Dq;
  float* ob = att + (size_t)b * Tq * Cq + h * Dq;

  for (int i = tid; i < Fq * Dq; i += 128) S[i] = 0.f;
  if (tid < Fq) zden[tid] = 0.f;
  __syncthreads();

  const int srw = tid >> 3, scc = (tid & 7) * 8;  // staging map: 16 rows x 64
  for (int c = 0; c < Tq / 16; ++c) {
    const int t0 = c * 16;

    // ---- stage Q', K', V chunk tiles (coalesced float4) -----------------
    {
      const float* qs = qpb + (size_t)(t0 + srw) * Fq;
      const float* ks = kpb + (size_t)(t0 + srw) * Fq;
      const float* vs = vb  + (size_t)(t0 + srw) * QKVC;
      *(float4*)&Qc[srw * LDQ + scc]     = *(const float4*)(qs + scc);
      *(float4*)&Qc[srw * LDQ + scc + 4] = *(const float4*)(qs + scc + 4);
      *(float4*)&Kc[srw * LDQ + scc]     = *(const float4*)(ks + scc);
      *(float4*)&Kc[srw * LDQ + scc + 4] = *(const float4*)(ks + scc + 4);
      *(float4*)&Vc[srw * LDQ + scc]     = *(const float4*)(vs + scc);
      *(float4*)&Vc[srw * LDQ + scc + 4] = *(const float4*)(vs + scc + 4);
    }
    __syncthreads();

    // ---- phase 1: wave 0 builds masked local scores M = tril(Q'K'^T) ----
    if (wave == 0) {
      v8f m = {};
      for (int k = 0; k < Fq; k += 4) {
        const int kk = k + 2 * half;
        v2f a  = *(const v2f*)&Qc[l16 * LDQ + kk];
        v2f bb = *(const v2f*)&Kc[l16 * LDQ + kk];   // B as (N,K)
        m = wmma4(a, bb, m);
      }
#pragma unroll
      for (int r = 0; r < 8; ++r) {
        const int mm = r + 8 * half;                 // row = t - t0
        Mloc[mm * LDM + l16] = (l16 <= mm) ? m[r] : 0.f;
      }
    }
    __syncthreads();

    // ---- phase 2: denominator (16 scalar lanes) + numerator tiles -------
    if (tid < 16) {
      float d = 1e-6f;
      const float* qrow = &Qc[tid * LDQ];
      for (int n = 0; n < 16; ++n) d += Mloc[tid * LDM + n];
      for (int f = 0; f < Fq; ++f) d += qrow[f] * zden[f];
      zden[Fq + tid] = d;
    }
    v8f acc = {};
    {
      // Q'(16x64) @ S_prev(64x64) : both operands from LDS
      for (int k = 0; k < Fq; k += 4) {
        const int kk = k + 2 * half;
        v2f a = *(const v2f*)&Qc[l16 * LDQ + kk];
        v2f bb; bb.x = S[kk * Dq + n0 + l16];
                bb.y = S[(kk + 1) * Dq + n0 + l16];
        acc = wmma4(a, bb, acc);
      }
      // M(16x16) @ V(16x64)
#pragma unroll
      for (int k = 0; k < 16; k += 4) {
        const int kk = k + 2 * half;
        v2f a = *(const v2f*)&Mloc[l16 * LDM + kk];
        v2f bb; bb.x = Vc[kk * LDQ + n0 + l16];
                bb.y = Vc[(kk + 1) * LDQ + n0 + l16];
        acc = wmma4(a, bb, acc);
      }
    }
    __syncthreads();

    // ---- phase 3: emit output, then rank-16 update of S and z -----------
#pragma unroll
    for (int r = 0; r < 8; ++r) {
      const int mm = r + 8 * half;
      ob[(size_t)(t0 + mm) * Cq + n0 + l16] = acc[r] / zden[Fq + mm];
    }
    for (int f0 = 0; f0 < Fq; f0 += 16) {
      v8f sacc = {};
#pragma unroll
      for (int k = 0; k < 16; k += 4) {
        const int kk = k + 2 * half;
        // A = K'^T fragment: A[f][t'] = Kc[t'][f]  (column gather, LDS)
        v2f a;  a.x = Kc[kk * LDQ + f0 + l16];
                a.y = Kc[(kk + 1) * LDQ + f0 + l16];
        v2f bb; bb.x = Vc[kk * LDQ + n0 + l16];
                bb.y = Vc[(kk + 1) * LDQ + n0 + l16];
        sacc = wmma4(a, bb, sacc);
      }
#pragma unroll
      for (int r = 0; r < 8; ++r)
        S[(f0 + r + 8 * half) * Dq + n0 + l16] += sacc[r];
    }
    if (tid < Fq) {
      float za = 0.f;
#pragma unroll
      for (int tp = 0; tp < 16; ++tp) za += Kc[tp * LDQ + tid];
      zden[tid] += za;
    }
    __syncthreads();   // also protects Qc/Kc/Vc against next chunk's staging
  }
}

// ---------------------------------------------------------------------------
// Host-side orchestration
// ---------------------------------------------------------------------------
extern "C" void kernel_launch(void* const* d_in, const int* in_sizes, int n_in,
                              void* d_out, int out_size, void* d_ws, size_t ws_size,
                              hipStream_t stream) {
  (void)in_sizes; (void)n_in; (void)out_size; (void)ws_size;
  const float* x      = (const float*)d_in[0];   // (B,T,C)
  const float* W_attn = (const float*)d_in[1];   // (3C,C)
  const float* W_proj = (const float*)d_in[2];   // (C,C)
  const float* proj   = (const float*)d_in[3];   // (D,F)
  float* out = (float*)d_out;                    // (B,T,C)

  // Workspace carve-up (floats), total ~38 MB
  float* ws  = (float*)d_ws;
  float* qkv = ws;                               // ROWS*QKVC = 4,718,592
  float* qp  = qkv + (size_t)ROWS * QKVC;        // BH*T*F    = 1,572,864
  float* kp  = qp  + (size_t)BH * Tq * Fq;       // BH*T*F    = 1,572,864
  float* att = kp  + (size_t)BH * Tq * Fq;       // ROWS*C    = 1,572,864

  // 1) qkv = x @ W_attn^T
  gemm_xwT_kernel<<<dim3(ROWS / 64, QKVC / 128), 256, 0, stream>>>(
      x, W_attn, qkv, ROWS, QKVC, Cq);
  // 2) phi features for q and k (includes row norms)
  feature_kernel<<<dim3(BH * (Tq / 16), 2), 128, 0, stream>>>(
      qkv, proj, qp, kp);
  // 3) chunked causal scan -> att in (B,T,C) layout
  scan_kernel<<<BH, 128, 0, stream>>>(qp, kp, qkv, att);
  // 4) out = att @ W_proj^T
  gemm_xwT_kernel<<<dim3(ROWS / 64, Cq / 128), 256, 0, stream>>>(
      att, W_proj, out, ROWS, Cq, Cq);
}